// Oscarmax_13211319402877
// MI455X (gfx1250) — compile-verified
//
#include <hip/hip_runtime.h>

#define DD 1024
#define NB 64
#define BETA_W 0.001f

typedef float v2f __attribute__((ext_vector_type(2)));
typedef float v8f __attribute__((ext_vector_type(8)));

__global__ __launch_bounds__(1024) void oscarmax_kernel(const float* __restrict__ x,
                                                        float* __restrict__ out) {
  __shared__ float sh_a[DD];     // |v|
  __shared__ float sh_s[DD];     // sorted |v| minus OSCAR weights
  __shared__ float sh_psum[DD];  // PAVA pool sums
  __shared__ int   sh_pcnt[DD];  // PAVA pool counts
  __shared__ float sh_pmean[DD]; // PAVA pool means
  __shared__ float sh_y[DD];     // isotonic result, clamped >= 0 (sorted order)
  __shared__ float sh_z[DD];     // signed, unsorted OSCAR prox output
  __shared__ float sh_zs[DD];    // z sorted descending
  __shared__ float sh_css[DD];   // cumsum of zs (WMMA partial, then fixed up)
  __shared__ float sh_off[64];   // per-group-of-16 scan offsets
  __shared__ float sh_redS[DD];  // reduction: sum of supported zs
  __shared__ float sh_redK[DD];  // reduction: support count

  const int row = blockIdx.x;
  const int i   = threadIdx.x;

  const float v   = x[row * DD + i];
  const float a   = fabsf(v);
  const float sgn = (v > 0.f) ? 1.f : ((v < 0.f) ? -1.f : 0.f);
  sh_a[i] = a;
  __syncthreads();

  // ---- rank sort of |v| descending, stable (matches argsort(-a)) ----
  int r1 = 0;
  for (int j = 0; j < DD; ++j) {
    float aj = sh_a[j];                       // LDS broadcast read
    r1 += (aj > a) || (aj == a && j < i);
  }
  // s[r] = a_sorted[r] - BETA*(D-1-r)
  sh_s[r1] = a - BETA_W * (float)(DD - 1 - r1);
  __syncthreads();

  // ---- PAVA nonincreasing isotonic regression (thread 0, O(D) amortized) ----
  if (i == 0) {
    int sp = 0;
    for (int t = 0; t < DD; ++t) {
      float cs = sh_s[t];
      int   cc = 1;
      float cm = cs;
      while (sp > 0 && sh_pmean[sp - 1] < cm) {   // violation of nonincrease -> merge
        --sp;
        cs += sh_psum[sp];
        cc += sh_pcnt[sp];
        cm = cs / (float)cc;
      }
      sh_psum[sp] = cs; sh_pcnt[sp] = cc; sh_pmean[sp] = cm; ++sp;
    }
    int pos = 0;
    for (int p = 0; p < sp; ++p) {
      float m = sh_pmean[p];
      if (m < 0.f) m = 0.f;                       // y = max(iso, 0)
      int c = sh_pcnt[p];
      for (int q = 0; q < c; ++q) sh_y[pos++] = m;
    }
  }
  __syncthreads();

  // ---- scatter back with signs ----
  const float z = sgn * sh_y[r1];
  sh_z[i] = z;
  __syncthreads();

  // ---- rank sort of z descending for sparsemax ----
  int r2 = 0;
  for (int j = 0; j < DD; ++j) {
    float zj = sh_z[j];
    r2 += (zj > z) || (zj == z && j < i);
  }
  sh_zs[r2] = z;
  __syncthreads();

  // ---- cumsum of zs via WMMA: within-16 prefix = S(16x16) x UpperTriOnes(16x16),
  //      K=16 done as 4 chained V_WMMA_F32_16X16X4_F32 (full f32 precision).
  //      Wave w (0..3) owns tile of 256 elements; EXEC all-ones per wave. ----
  const int wv = i >> 5;
  if (wv < 4) {
    const int lane = i & 31;
    const int half = lane >> 4;   // 0: K comps {0,1}; 1: K comps {2,3}
    const int l16  = lane & 15;
    const int base = 256 * wv;
    v8f c = {};
#pragma unroll
    for (int t = 0; t < 4; ++t) {
      const int k0 = 4 * t + 2 * half;
      v2f av, bv;
      // A (16x4 f32): lanes0-15 M=l16 K={4t,4t+1}; lanes16-31 M=l16 K={4t+2,4t+3}
      av.x = sh_zs[base + 16 * l16 + k0];
      av.y = sh_zs[base + 16 * l16 + k0 + 1];
      // B (4x16 f32) = U, U[k][n] = (k <= n); N=l16 per lane, K mirrors A halves
      bv.x = (k0 <= l16) ? 1.f : 0.f;
      bv.y = (k0 + 1 <= l16) ? 1.f : 0.f;
      c = __builtin_amdgcn_wmma_f32_16x16x4_f32(false, av, false, bv,
                                                (short)0, c, false, false);
    }
    // D (16x16 f32): VGPR g -> M=g (lanes0-15) / M=g+8 (lanes16-31), N=l16
#pragma unroll
    for (int g = 0; g < 8; ++g) {
      const int m = g + 8 * half;
      sh_css[base + 16 * m + l16] = c[g];
    }
  }
  __syncthreads();

  // group-of-16 offsets (64 groups, sequential — trivial)
  if (i == 0) {
    float run = 0.f;
    for (int grp = 0; grp < 64; ++grp) {
      sh_off[grp] = run;
      run += sh_css[grp * 16 + 15];   // group total = last within-group prefix
    }
  }
  __syncthreads();

  // ---- sparsemax threshold ----
  const float css_i = sh_css[i] + sh_off[i >> 4];
  const float zs_i  = sh_zs[i];
  const float flag  = (1.f + (float)(i + 1) * zs_i > css_i) ? 1.f : 0.f;
  sh_redS[i] = flag * zs_i;
  sh_redK[i] = flag;
  __syncthreads();
  for (int st = 512; st > 0; st >>= 1) {
    if (i < st) {
      sh_redS[i] += sh_redS[i + st];
      sh_redK[i] += sh_redK[i + st];
    }
    __syncthreads();
  }
  const float S   = sh_redS[0];
  const float K   = sh_redK[0];
  const float tau = (S - 1.f) / K;

  float o = z - tau;
  out[row * DD + i] = (o > 0.f) ? o : 0.f;
}

extern "C" void kernel_launch(void* const* d_in, const int* in_sizes, int n_in,
                              void* d_out, int out_size, void* d_ws, size_t ws_size,
                              hipStream_t stream) {
  (void)in_sizes; (void)n_in; (void)out_size; (void)d_ws; (void)ws_size;
  const float* x   = (const float*)d_in[0];
  float*       out = (float*)d_out;
  oscarmax_kernel<<<NB, 1024, 0, stream>>>(x, out);
}